// Model4GCN_1889785610423
// MI455X (gfx1250) — compile-verified
//
#include <hip/hip_runtime.h>
#include <cmath>

#define NN 20000
#define TT 8
#define HH 128
#define EE 320000
#define ETOT (EE + NN)      // 340000
#define NSTRIPS (NN / 16)   // 1250 (exact)

typedef __attribute__((ext_vector_type(2))) float v2f;
typedef __attribute__((ext_vector_type(8))) float v8f;

// ---------------- graph prep ----------------
__global__ void k_deg_init(float* deg) {
  int i = blockIdx.x * blockDim.x + threadIdx.x;
  if (i < NN) deg[i] = 1.0f;  // self loop contributes 1 per node
}

__global__ void k_edges(const int* __restrict__ ei, int* __restrict__ src32,
                        int* __restrict__ dst32, float* __restrict__ deg) {
  int e = blockIdx.x * blockDim.x + threadIdx.x;
  if (e >= ETOT) return;
  if (e < EE) {
    int s = ei[e];
    int d = ei[EE + e];
    src32[e] = s;
    dst32[e] = d;
    unsafeAtomicAdd(&deg[d], 1.0f);
  } else {
    int i = e - EE;  // self loop
    src32[e] = i;
    dst32[e] = i;
  }
}

__global__ void k_dinv(float* deg) {
  int i = blockIdx.x * blockDim.x + threadIdx.x;
  if (i < NN) {
    float d = deg[i];
    deg[i] = (d > 0.0f) ? (1.0f / sqrtf(d)) : 0.0f;  // in-place deg -> d^-1/2
  }
}

__global__ void k_norm(const int* __restrict__ src32, const int* __restrict__ dst32,
                       const float* __restrict__ dinv, float* __restrict__ norm) {
  int e = blockIdx.x * blockDim.x + threadIdx.x;
  if (e < ETOT) norm[e] = dinv[src32[e]] * dinv[dst32[e]];
}

__global__ void k_fill(float* __restrict__ p, float v, int n) {
  int i = blockIdx.x * blockDim.x + threadIdx.x;
  if (i < n) p[i] = v;
}

__global__ void k_fill_ptr(float* __restrict__ p, const float* __restrict__ v, int n) {
  int i = blockIdx.x * blockDim.x + threadIdx.x;
  if (i < n) p[i] = v[0];
}

// ---------------- scatter-add aggregations ----------------
// Layer 1: F_IN==1, so agg(x*Wc1) == (agg x) * Wc1 -> scalar scatter over all T at once.
__global__ void k_scatter_s(const float* __restrict__ x, const int* __restrict__ src32,
                            const int* __restrict__ dst32, const float* __restrict__ norm,
                            float* __restrict__ aggs) {
  int e = blockIdx.x * blockDim.x + threadIdx.x;
  if (e >= ETOT) return;
  int s = src32[e], d = dst32[e];
  float w = norm[e];
  const float4* xs = (const float4*)(x + (size_t)s * TT);
  float4 a = xs[0], b = xs[1];
  float* o = aggs + (size_t)d * TT;
  unsafeAtomicAdd(o + 0, a.x * w);
  unsafeAtomicAdd(o + 1, a.y * w);
  unsafeAtomicAdd(o + 2, a.z * w);
  unsafeAtomicAdd(o + 3, a.w * w);
  unsafeAtomicAdd(o + 4, b.x * w);
  unsafeAtomicAdd(o + 5, b.y * w);
  unsafeAtomicAdd(o + 6, b.z * w);
  unsafeAtomicAdd(o + 7, b.w * w);
}

// Layers 2/3: 128-wide feature scatter; one 32-lane group per edge, float4 per lane.
__global__ void k_scatter_v(const float* __restrict__ xw, const int* __restrict__ src32,
                            const int* __restrict__ dst32, const float* __restrict__ norm,
                            float* __restrict__ agg) {
  int gid = blockIdx.x * blockDim.x + threadIdx.x;
  int e = gid >> 5, lane = gid & 31;
  if (e >= ETOT) return;
  int s = src32[e], d = dst32[e];
  float w = norm[e];
  float4 v = ((const float4*)(xw + (size_t)s * HH))[lane];
  float* o = agg + (size_t)d * HH + lane * 4;
  unsafeAtomicAdd(o + 0, v.x * w);
  unsafeAtomicAdd(o + 1, v.y * w);
  unsafeAtomicAdd(o + 2, v.z * w);
  unsafeAtomicAdd(o + 3, v.w * w);
}

__global__ void k_scatter_out(const float* __restrict__ yw, const int* __restrict__ src32,
                              const int* __restrict__ dst32, const float* __restrict__ norm,
                              float* __restrict__ out) {
  int e = blockIdx.x * blockDim.x + threadIdx.x;
  if (e >= ETOT) return;
  int s = src32[e], d = dst32[e];
  float w = norm[e];
  const float4* ys = (const float4*)(yw + (size_t)s * TT);
  float4 a = ys[0], b = ys[1];
  float* o = out + (size_t)d * TT;
  unsafeAtomicAdd(o + 0, a.x * w);
  unsafeAtomicAdd(o + 1, a.y * w);
  unsafeAtomicAdd(o + 2, a.z * w);
  unsafeAtomicAdd(o + 3, a.w * w);
  unsafeAtomicAdd(o + 4, b.x * w);
  unsafeAtomicAdd(o + 5, b.y * w);
  unsafeAtomicAdd(o + 6, b.z * w);
  unsafeAtomicAdd(o + 7, b.w * w);
}

// ---------------- WMMA fp32 GEMMs ----------------
// xw_t = h[:, t, :] @ Wc   (M=20000, K=128, N=128), B staged in LDS.
__global__ __launch_bounds__(256) void k_gemm_xw(const float* __restrict__ Hbuf,
                                                 const float* __restrict__ Wc,
                                                 float* __restrict__ xw, int t) {
  __shared__ float Bs[HH * HH];
  int tid = threadIdx.x;
  {
    const float4* B4 = (const float4*)Wc;
    float4* Bs4 = (float4*)Bs;
    for (int i = tid; i < HH * HH / 4; i += 256) Bs4[i] = B4[i];
  }
  __syncthreads();
  int wave = tid >> 5, lane = tid & 31;
  int strip = blockIdx.x * 8 + wave;
  if (strip >= NSTRIPS) return;  // wave-uniform; EXEC stays all-1s for WMMA
  int col = lane & 15;
  int khalf = (lane >> 4) << 1;  // lanes 0-15: K pair 0,1 ; lanes 16-31: K pair 2,3
  const float* Arow = Hbuf + ((size_t)(strip * 16 + col) * TT + t) * HH;
  v8f acc[8] = {};
  for (int k0 = 0; k0 < HH; k0 += 4) {
    v2f a;
    a.x = Arow[k0 + khalf];
    a.y = Arow[k0 + khalf + 1];
    const float* b0 = &Bs[(k0 + khalf) * HH];
    const float* b1 = &Bs[(k0 + khalf + 1) * HH];
#pragma unroll
    for (int n = 0; n < 8; ++n) {
      v2f b;
      b.x = b0[n * 16 + col];
      b.y = b1[n * 16 + col];
      acc[n] = __builtin_amdgcn_wmma_f32_16x16x4_f32(false, a, false, b, (short)0,
                                                     acc[n], false, false);
    }
  }
  int rbase = strip * 16 + 8 * (lane >> 4);
#pragma unroll
  for (int n = 0; n < 8; ++n) {
#pragma unroll
    for (int j = 0; j < 8; ++j) {
      xw[(size_t)(rbase + j) * HH + n * 16 + col] = acc[n][j];
    }
  }
}

// h[:, t, :] = tanh(i2h + h[:, t-1, :] @ Wl + bl)
// mode 0 (layer 1): i2h = aggS[n,t] * wc1[col] + bc[col]
// mode 1 (layers 2/3): i2h = aggV[n,col] + bc[col]
__global__ __launch_bounds__(256) void k_gemm_rnn(
    float* __restrict__ Hbuf, const float* __restrict__ Wl,
    const float* __restrict__ aggV, const float* __restrict__ aggS,
    const float* __restrict__ wc1, const float* __restrict__ bc,
    const float* __restrict__ bl, int t, int has_prev, int mode) {
  __shared__ float Bs[HH * HH];
  int tid = threadIdx.x;
  if (has_prev) {
    const float4* B4 = (const float4*)Wl;
    float4* Bs4 = (float4*)Bs;
    for (int i = tid; i < HH * HH / 4; i += 256) Bs4[i] = B4[i];
    __syncthreads();
  }
  int wave = tid >> 5, lane = tid & 31;
  int strip = blockIdx.x * 8 + wave;
  if (strip >= NSTRIPS) return;
  int col = lane & 15;
  int khalf = (lane >> 4) << 1;
  v8f acc[8] = {};
  if (has_prev) {
    const float* Arow = Hbuf + ((size_t)(strip * 16 + col) * TT + (t - 1)) * HH;
    for (int k0 = 0; k0 < HH; k0 += 4) {
      v2f a;
      a.x = Arow[k0 + khalf];
      a.y = Arow[k0 + khalf + 1];
      const float* b0 = &Bs[(k0 + khalf) * HH];
      const float* b1 = &Bs[(k0 + khalf + 1) * HH];
#pragma unroll
      for (int n = 0; n < 8; ++n) {
        v2f b;
        b.x = b0[n * 16 + col];
        b.y = b1[n * 16 + col];
        acc[n] = __builtin_amdgcn_wmma_f32_16x16x4_f32(false, a, false, b, (short)0,
                                                       acc[n], false, false);
      }
    }
  }
  int rbase = strip * 16 + 8 * (lane >> 4);
  float bsum[8], wcv[8];
#pragma unroll
  for (int n = 0; n < 8; ++n) {
    int cg = n * 16 + col;
    bsum[n] = bc[cg] + bl[cg];
    wcv[n] = (mode == 0) ? wc1[cg] : 0.0f;
  }
  float sag[8] = {};
  if (mode == 0) {
#pragma unroll
    for (int j = 0; j < 8; ++j) sag[j] = aggS[(size_t)(rbase + j) * TT + t];
  }
#pragma unroll
  for (int n = 0; n < 8; ++n) {
#pragma unroll
    for (int j = 0; j < 8; ++j) {
      int r = rbase + j;
      int cg = n * 16 + col;
      float v = acc[n][j] + bsum[n];
      if (mode == 0) v += sag[j] * wcv[n];
      else           v += aggV[(size_t)r * HH + cg];
      Hbuf[((size_t)r * TT + t) * HH + cg] = tanhf(v);
    }
  }
}

// yw[n,t] = dot(h[n,t,:], Wco)
__global__ void k_yw(const float* __restrict__ h, const float* __restrict__ Wco,
                     float* __restrict__ yw) {
  int i = blockIdx.x * blockDim.x + threadIdx.x;
  if (i >= NN * TT) return;
  const float4* hr = (const float4*)(h + (size_t)i * HH);
  const float4* w4 = (const float4*)Wco;
  float acc = 0.0f;
#pragma unroll
  for (int k = 0; k < HH / 4; ++k) {
    float4 a = hr[k], b = w4[k];
    acc += a.x * b.x + a.y * b.y + a.z * b.z + a.w * b.w;
  }
  yw[i] = acc;
}

// ---------------- host side ----------------
extern "C" void kernel_launch(void* const* d_in, const int* in_sizes, int n_in,
                              void* d_out, int out_size, void* d_ws, size_t ws_size,
                              hipStream_t stream) {
  (void)in_sizes; (void)n_in; (void)out_size; (void)ws_size;
  const float* x   = (const float*)d_in[0];
  const int*   ei  = (const int*)d_in[1];
  const float* Wc1 = (const float*)d_in[2];
  const float* bc1 = (const float*)d_in[3];
  const float* Wl1 = (const float*)d_in[4];
  const float* bl1 = (const float*)d_in[5];
  const float* Wc2 = (const float*)d_in[6];
  const float* bc2 = (const float*)d_in[7];
  const float* Wl2 = (const float*)d_in[8];
  const float* bl2 = (const float*)d_in[9];
  const float* Wc3 = (const float*)d_in[10];
  const float* bc3 = (const float*)d_in[11];
  const float* Wl3 = (const float*)d_in[12];
  const float* bl3 = (const float*)d_in[13];
  const float* Wco = (const float*)d_in[14];
  const float* bco = (const float*)d_in[15];
  float* out = (float*)d_out;

  char* ws = (char*)d_ws;
  auto carve = [&](size_t bytes) -> void* {
    void* p = (void*)ws;
    ws += (bytes + 255) & ~(size_t)255;
    return p;
  };
  float* deg   = (float*)carve((size_t)NN * 4);            // becomes dinv in-place
  int*   src32 = (int*)carve((size_t)ETOT * 4);
  int*   dst32 = (int*)carve((size_t)ETOT * 4);
  float* norm  = (float*)carve((size_t)ETOT * 4);
  float* aggs  = (float*)carve((size_t)NN * TT * 4);
  float* agg   = (float*)carve((size_t)NN * HH * 4);
  float* xw    = (float*)carve((size_t)NN * HH * 4);
  float* yw    = (float*)carve((size_t)NN * TT * 4);
  float* h     = (float*)carve((size_t)NN * TT * HH * 4);  // in-place across layers

  const dim3 B(256);
  const int gemmBlocks = (NSTRIPS + 7) / 8;  // 157

  // graph prep
  k_deg_init<<<(NN + 255) / 256, B, 0, stream>>>(deg);
  k_edges<<<(ETOT + 255) / 256, B, 0, stream>>>(ei, src32, dst32, deg);
  k_dinv<<<(NN + 255) / 256, B, 0, stream>>>(deg);
  k_norm<<<(ETOT + 255) / 256, B, 0, stream>>>(src32, dst32, deg, norm);

  // ---- layer 1 (scalar aggregation, all T at once) ----
  k_fill<<<(NN * TT + 255) / 256, B, 0, stream>>>(aggs, 0.0f, NN * TT);
  k_scatter_s<<<(ETOT + 255) / 256, B, 0, stream>>>(x, src32, dst32, norm, aggs);
  for (int t = 0; t < TT; ++t) {
    k_gemm_rnn<<<gemmBlocks, B, 0, stream>>>(h, Wl1, nullptr, aggs, Wc1, bc1, bl1,
                                             t, t > 0 ? 1 : 0, 0);
  }

  // ---- layers 2 and 3 (in-place over h) ----
  const float* Wcs[2] = {Wc2, Wc3};
  const float* bcs[2] = {bc2, bc3};
  const float* Wls[2] = {Wl2, Wl3};
  const float* bls[2] = {bl2, bl3};
  for (int l = 0; l < 2; ++l) {
    for (int t = 0; t < TT; ++t) {
      k_gemm_xw<<<gemmBlocks, B, 0, stream>>>(h, Wcs[l], xw, t);
      k_fill<<<(NN * HH + 255) / 256, B, 0, stream>>>(agg, 0.0f, NN * HH);
      k_scatter_v<<<(ETOT * 32 + 255) / 256, B, 0, stream>>>(xw, src32, dst32, norm, agg);
      k_gemm_rnn<<<gemmBlocks, B, 0, stream>>>(h, Wls[l], agg, nullptr, nullptr,
                                               bcs[l], bls[l], t, t > 0 ? 1 : 0, 1);
    }
  }

  // ---- output GCN (scalar features) ----
  k_yw<<<(NN * TT + 255) / 256, B, 0, stream>>>(h, Wco, yw);
  k_fill_ptr<<<(NN * TT + 255) / 256, B, 0, stream>>>(out, bco, NN * TT);
  k_scatter_out<<<(ETOT + 255) / 256, B, 0, stream>>>(yw, src32, dst32, norm, out);
}